// MaskLoss_59210419143319
// MI455X (gfx1250) — compile-verified
//
#include <hip/hip_runtime.h>
#include <stdint.h>

#define B_          256
#define HW_         4096                  // 64*64
#define CH_STRIDE   135168                // 33*4096
#define MAIN_ELEMS  131072                // 32*4096
#define TPB         256
#define NWAVE       (TPB / 32)
#define VEC         4
#define ITERS       (MAIN_ELEMS / (TPB * VEC))   // 128
#define STAGES      8                     // 64 KB LDS, ~14 MB in flight grid-wide
#define NEG_HUGE    (-3.402823466e+38f)

// ---------------- CDNA5 async global->LDS path (guarded) ----------------
typedef int v4i __attribute__((ext_vector_type(4)));
typedef __attribute__((address_space(1))) v4i as1_v4i;   // global 16B packet
typedef __attribute__((address_space(3))) v4i as3_v4i;   // LDS 16B packet

#if defined(__has_builtin)
#if __has_builtin(__builtin_amdgcn_global_load_async_to_lds_b128)
#define HAVE_ASYNC_LDS 1
#endif
#endif
#ifndef HAVE_ASYNC_LDS
#define HAVE_ASYNC_LDS 0
#endif

#if defined(__has_builtin) && __has_builtin(__builtin_amdgcn_s_wait_asynccnt)
#define WAIT_ASYNC(n) __builtin_amdgcn_s_wait_asynccnt(n)
#else
#define WAIT_ASYNC(n) asm volatile("s_wait_asynccnt %0" ::"i"(n) : "memory")
#endif

// One workgroup per batch. 8 wave32s.
__global__ __launch_bounds__(TPB) void maskloss_batch(
    const float* __restrict__ pred, const float* __restrict__ targ,
    float* __restrict__ ws_loss, float* __restrict__ ws_cnt)
{
    const int b    = blockIdx.x;
    const int tid  = threadIdx.x;
    const int lane = tid & 31;
    const int wave = tid >> 5;
    const size_t base = (size_t)b * CH_STRIDE;

    __shared__ float s_red[NWAVE * 5];
    __shared__ float s_bcast[2];
#if HAVE_ASYNC_LDS
    __shared__ __align__(16) float s_bp[STAGES][TPB * VEC];
    __shared__ __align__(16) float s_bt[STAGES][TPB * VEC];
#endif

    // ---------------- Phase A: grow-channel (c==32) stats -----------------
    {
        const float* tg = targ + base + MAIN_ELEMS;
        const float* pg = pred + base + MAIN_ELEMS;
        float lsum = 0.f, psum = 0.f, lcnt = 0.f, pcnt = 0.f;
        for (int i = tid * VEC; i < HW_; i += TPB * VEC) {
            float4 t4 = *(const float4*)(tg + i);
            float4 p4 = *(const float4*)(pg + i);
            const float tv[4] = {t4.x, t4.y, t4.z, t4.w};
            const float pv[4] = {p4.x, p4.y, p4.z, p4.w};
#pragma unroll
            for (int k = 0; k < 4; ++k) {
                const bool gm = (tv[k] != 0.0f);           // grow_mask
                lsum += tv[k];                              // zeros add exactly 0
                lcnt += gm ? 1.f : 0.f;
                psum += gm ? pv[k] : 0.f;                   // pred_grow = pred*mask
                pcnt += (gm && pv[k] != 0.0f) ? 1.f : 0.f;  // count_nonzero(pred_grow)
            }
        }
#pragma unroll
        for (int off = 16; off; off >>= 1) {
            lsum += __shfl_down(lsum, off, 32);
            lcnt += __shfl_down(lcnt, off, 32);
            psum += __shfl_down(psum, off, 32);
            pcnt += __shfl_down(pcnt, off, 32);
        }
        if (lane == 0) {
            s_red[wave]             = lsum;
            s_red[NWAVE + wave]     = lcnt;
            s_red[2 * NWAVE + wave] = psum;
            s_red[3 * NWAVE + wave] = pcnt;
        }
        __syncthreads();
        if (tid == 0) {
            float LS = 0.f, LC = 0.f, PS = 0.f, PC = 0.f;
            for (int w = 0; w < NWAVE; ++w) {
                LS += s_red[w];
                LC += s_red[NWAVE + w];
                PS += s_red[2 * NWAVE + w];
                PC += s_red[3 * NWAVE + w];
            }
            s_bcast[0] = (LC != 0.f) ? (LS / LC) : 0.f;  // label_mean
            s_bcast[1] = (PC != 0.f) ? (PS / PC) : 0.f;  // pred_mean
        }
        __syncthreads();
    }

    // ------------- Phase B: main channels, single-pass online LSE ----------
    float m = NEG_HUGE, s = 0.f;       // running (max, sum exp(x-m))
    float lx = 0.f, lt = 0.f, cnt = 0.f;
    const float* pm = pred + base;
    const float* tm = targ + base;

    // Branchless online logsumexp + masked sums. A wave32 with a 50% random
    // mask essentially never has EXEC==0, so EXEC-branching only costs SALU.
    auto accum = [&](float t, float x) {
        const bool valid = (t != 0.0f);                 // main_mask
        cnt += valid ? 1.f : 0.f;
        lt  += t;                                       // masked-out t is exactly 0
        lx  += t * x;
        const float xm = valid ? x : NEG_HUGE;
        const float Mn = fmaxf(m, xm);
        const float e  = valid ? __expf(xm - Mn) : 0.f; // gate: empty-acc edge case
        s = s * __expf(m - Mn) + e;                     // m=-HUGE => exp -> 0, s=0
        m = Mn;
    };

#if HAVE_ASYNC_LDS
    // 8-stage async global->LDS pipeline; each lane fetches/consumes its own 16B.
    auto issue = [&](int i) {
        const int bsel  = i & (STAGES - 1);
        const float* gp = pm + (size_t)i * TPB * VEC + tid * VEC;
        const float* gt = tm + (size_t)i * TPB * VEC + tid * VEC;
        __builtin_amdgcn_global_load_async_to_lds_b128(
            (as1_v4i*)gp, (as3_v4i*)&s_bp[bsel][tid * VEC], 0, 0);
        __builtin_amdgcn_global_load_async_to_lds_b128(
            (as1_v4i*)gt, (as3_v4i*)&s_bt[bsel][tid * VEC], 0, 0);
    };
    auto consume = [&](int i) {
        const int bsel = i & (STAGES - 1);
        float4 t4 = *(const float4*)&s_bt[bsel][tid * VEC];
        float4 p4 = *(const float4*)&s_bp[bsel][tid * VEC];
        accum(t4.x, p4.x); accum(t4.y, p4.y);
        accum(t4.z, p4.z); accum(t4.w, p4.w);
    };

    for (int k = 0; k < STAGES - 1 && k < ITERS; ++k) issue(k);
    int i = 0;
    // steady state: constant wait (async loads complete in order)
    for (; i + (STAGES - 1) < ITERS; ++i) {
        issue(i + STAGES - 1);
        WAIT_ASYNC(2 * (STAGES - 1));   // 14: everything older than iter i done
        consume(i);
    }
    // drain: decreasing number of younger outstanding iterations
    for (; i < ITERS; ++i) {
        switch (ITERS - 1 - i) {
            case 6: WAIT_ASYNC(12); break;
            case 5: WAIT_ASYNC(10); break;
            case 4: WAIT_ASYNC(8);  break;
            case 3: WAIT_ASYNC(6);  break;
            case 2: WAIT_ASYNC(4);  break;
            case 1: WAIT_ASYNC(2);  break;
            default: WAIT_ASYNC(0); break;
        }
        consume(i);
    }
#else
    for (int i = 0; i < ITERS; ++i) {
        const int idx = i * TPB * VEC + tid * VEC;
        if (i + 8 < ITERS) {           // -> global_prefetch_b8
            __builtin_prefetch(pm + idx + 8 * TPB * VEC, 0, 0);
            __builtin_prefetch(tm + idx + 8 * TPB * VEC, 0, 0);
        }
        float4 t4 = *(const float4*)(tm + idx);
        float4 p4 = *(const float4*)(pm + idx);
        accum(t4.x, p4.x); accum(t4.y, p4.y);
        accum(t4.z, p4.z); accum(t4.w, p4.w);
    }
#endif

    // ---- merge (m,s) pairs + sums: wave32 shuffle, then across 8 waves ----
#pragma unroll
    for (int off = 16; off; off >>= 1) {
        float om = __shfl_down(m, off, 32);
        float os = __shfl_down(s, off, 32);
        float M  = fmaxf(m, om);
        s = s * __expf(m - M) + os * __expf(om - M);  // -HUGE lanes underflow to 0
        m = M;
        lx  += __shfl_down(lx, off, 32);
        lt  += __shfl_down(lt, off, 32);
        cnt += __shfl_down(cnt, off, 32);
    }
    if (lane == 0) {
        s_red[wave]             = m;
        s_red[NWAVE + wave]     = s;
        s_red[2 * NWAVE + wave] = lx;
        s_red[3 * NWAVE + wave] = lt;
        s_red[4 * NWAVE + wave] = cnt;
    }
    __syncthreads();
    if (tid == 0) {
        float M  = s_red[0], S = s_red[NWAVE];
        float LX = s_red[2 * NWAVE], LT = s_red[3 * NWAVE], CN = s_red[4 * NWAVE];
        for (int w = 1; w < NWAVE; ++w) {
            float om = s_red[w], os = s_red[NWAVE + w];
            float Mn = fmaxf(M, om);
            S = S * __expf(M - Mn) + os * __expf(om - Mn);
            M = Mn;
            LX += s_red[2 * NWAVE + w];
            LT += s_red[3 * NWAVE + w];
            CN += s_red[4 * NWAVE + w];
        }
        // fold in the always-valid extra element x = pred_mean, label = label_mean
        const float label_mean = s_bcast[0];
        const float pred_mean  = s_bcast[1];
        float Mn = fmaxf(M, pred_mean);
        S = S * __expf(M - Mn) + __expf(pred_mean - Mn);
        M = Mn;
        const float lse = M + logf(S);
        LX += label_mean * pred_mean;
        LT += label_mean;
        ws_loss[b] = lse * LT - LX;   // = -sum(label * (x - lse))
        ws_cnt[b]  = CN + 1.0f;
    }
}

// Deterministic tree-sum of 256 per-batch partials; out = loss/count.
__global__ __launch_bounds__(B_) void maskloss_final(
    const float* __restrict__ ws_loss, const float* __restrict__ ws_cnt,
    float* __restrict__ out)
{
    __shared__ float sl[B_];
    __shared__ float sc[B_];
    const int t = threadIdx.x;
    sl[t] = ws_loss[t];
    sc[t] = ws_cnt[t];
    __syncthreads();
    for (int off = B_ / 2; off; off >>= 1) {
        if (t < off) { sl[t] += sl[t + off]; sc[t] += sc[t + off]; }
        __syncthreads();
    }
    if (t == 0) out[0] = sl[0] / sc[0];
}

extern "C" void kernel_launch(void* const* d_in, const int* in_sizes, int n_in,
                              void* d_out, int out_size, void* d_ws, size_t ws_size,
                              hipStream_t stream)
{
    (void)in_sizes; (void)n_in; (void)out_size; (void)ws_size;
    const float* pred = (const float*)d_in[0];   // predictions (256,33,64,64) f32
    const float* targ = (const float*)d_in[1];   // targets     (256,33,64,64) f32
    float* ws_loss = (float*)d_ws;               // 256 floats
    float* ws_cnt  = ws_loss + B_;               // 256 floats

    maskloss_batch<<<B_, TPB, 0, stream>>>(pred, targ, ws_loss, ws_cnt);
    maskloss_final<<<1, B_, 0, stream>>>(ws_loss, ws_cnt, (float*)d_out);
}